// DensityField_28492813042010
// MI455X (gfx1250) — compile-verified
//
#include <hip/hip_runtime.h>
#include <math.h>

// ---------------------------------------------------------------------------
// CDNA5 / gfx1250 fused Instant-NGP density field
//   Kernel 1: hash-grid encode (gathers, table resident in 192MB L2)
//             + 5 chained GEMM stages on V_WMMA_F32_16X16X4_F32 (fp32 WMMA),
//             activations staged in LDS per 16-point tile, never hit HBM.
//   Kernel 2: per-ray volume-rendering weights (wave32 prefix scan).
// ---------------------------------------------------------------------------

typedef float v2f __attribute__((ext_vector_type(2)));
typedef float v8f __attribute__((ext_vector_type(8)));

#define HASH_SIZE (1u << 19)
#define WAVES_PER_BLOCK 8

static __device__ inline v8f zero8() {
    v8f z;
#pragma unroll
    for (int i = 0; i < 8; ++i) z[i] = 0.0f;
    return z;
}

// D = A(16x4,f32) * B(4x16,f32) + C(16x16,f32)
static __device__ inline v8f wmma4(v2f a, v2f b, v8f c) {
    return __builtin_amdgcn_wmma_f32_16x16x4_f32(
        /*neg_a=*/false, a, /*neg_b=*/false, b,
        /*c_mod=*/(short)0, c, /*reuse_a=*/false, /*reuse_b=*/false);
}

__global__ __launch_bounds__(256) void ngp_fused_kernel(
    const float* __restrict__ xyz,     // (N,3)
    const float* __restrict__ table,   // (16, 2^19, 2)
    const float* __restrict__ w_in,    // (32,64)
    const float* __restrict__ w_out,   // (64,16)
    const float* __restrict__ rgb_w1,  // (15,64)
    const float* __restrict__ rgb_w2,  // (64,64)
    const float* __restrict__ rgb_w3,  // (64,3)
    float* __restrict__ out,           // (N,4): [logdens placeholder, r,g,b]
    int nPoints)
{
    // per-wave tiles: enc 16x32, activation stage 16x64
    __shared__ float encS[WAVES_PER_BLOCK][16][32];
    __shared__ float stage[WAVES_PER_BLOCK][16][64];

    const int tid  = threadIdx.x;
    const int wave = tid >> 5;
    const int lane = tid & 31;
    const int q    = lane & 15;   // point within tile (A/B lane index)
    const int half = lane >> 4;   // K-half selector per ISA layout
    const int tileBase = (blockIdx.x * WAVES_PER_BLOCK + wave) * 16;

    // ---------------- stage 0: hash-grid encoding -------------------------
    // lane (q, half) computes levels [8*half, 8*half+8) of point q
    {
        const int p = tileBase + q;
        const float px = xyz[(size_t)p * 3 + 0];
        const float py = xyz[(size_t)p * 3 + 1];
        const float pz = xyz[(size_t)p * 3 + 2];
        const float RES[16] = {16.f, 23.f, 33.f, 48.f, 70.f, 101.f, 147.f, 212.f,
                               307.f, 445.f, 645.f, 933.f, 1351.f, 1955.f, 2830.f, 4096.f};
        const int lvl0 = half * 8;
#pragma unroll
        for (int li = 0; li < 8; ++li) {
            const int lvl = lvl0 + li;
            const float r = RES[lvl];
            const float fx = px * r, fy = py * r, fz = pz * r;
            const float x0 = floorf(fx), y0 = floorf(fy), z0 = floorf(fz);
            const float dx = fx - x0, dy = fy - y0, dz = fz - z0;
            const unsigned ux = (unsigned)x0, uy = (unsigned)y0, uz = (unsigned)z0;
            const float* tab = table + (size_t)lvl * (size_t)HASH_SIZE * 2u;
            float f0 = 0.0f, f1 = 0.0f;
#pragma unroll
            for (int c = 0; c < 8; ++c) {
                const unsigned bx = (c >> 2) & 1u, by = (c >> 1) & 1u, bz = c & 1u;
                const unsigned h = (ux + bx) * 1u
                                 ^ (uy + by) * 2654435761u
                                 ^ (uz + bz) * 805459861u;
                const unsigned idx = h & (HASH_SIZE - 1u);
                const float w = (bx ? dx : 1.0f - dx)
                              * (by ? dy : 1.0f - dy)
                              * (bz ? dz : 1.0f - dz);
                const float2 t = *(const float2*)(tab + (size_t)idx * 2u);
                f0 += w * t.x;
                f1 += w * t.y;
            }
            encS[wave][q][2 * lvl + 0] = f0;
            encS[wave][q][2 * lvl + 1] = f1;
        }
    }
    __syncthreads();

    // ---------------- stage 1: h = relu(enc(16x32) @ w_in(32x64)) ---------
    v8f h4[4];
#pragma unroll
    for (int nt = 0; nt < 4; ++nt) h4[nt] = zero8();
#pragma unroll
    for (int ks = 0; ks < 8; ++ks) {
        const int kk = ks * 4 + half * 2;
        v2f a; a.x = encS[wave][q][kk]; a.y = encS[wave][q][kk + 1];
#pragma unroll
        for (int nt = 0; nt < 4; ++nt) {
            v2f b;
            b.x = w_in[kk * 64 + nt * 16 + q];
            b.y = w_in[(kk + 1) * 64 + nt * 16 + q];
            h4[nt] = wmma4(a, b, h4[nt]);
        }
    }
#pragma unroll
    for (int nt = 0; nt < 4; ++nt)
#pragma unroll
        for (int i = 0; i < 8; ++i)
            stage[wave][i + half * 8][nt * 16 + q] = fmaxf(h4[nt][i], 0.0f);
    __syncthreads();

    // ---------------- stage 2: raw = h(16x64) @ w_out(64x16) --------------
    v8f raw = zero8();
#pragma unroll
    for (int ks = 0; ks < 16; ++ks) {
        const int kk = ks * 4 + half * 2;
        v2f a; a.x = stage[wave][q][kk]; a.y = stage[wave][q][kk + 1];
        v2f b;
        b.x = w_out[kk * 16 + q];
        b.y = w_out[(kk + 1) * 16 + q];
        raw = wmma4(a, b, raw);
    }
    __syncthreads();

    // log-density = raw[:,0] -> out channel 0 (kernel 2 turns it into weights)
    if (q == 0) {
#pragma unroll
        for (int i = 0; i < 8; ++i)
            out[(size_t)(tileBase + i + half * 8) * 4 + 0] = raw[i];
    }
    // feats = raw[:,1:16] shifted into stage[.][0..14], col 15 zero-padded
#pragma unroll
    for (int i = 0; i < 8; ++i) {
        const int m = i + half * 8;
        if (q >= 1) stage[wave][m][q - 1] = raw[i];
        else        stage[wave][m][15]    = 0.0f;
    }
    __syncthreads();

    // ---------------- stage 3: h1 = relu(feats(16x16) @ rgb_w1(15x64)) ----
    v8f h1[4];
#pragma unroll
    for (int nt = 0; nt < 4; ++nt) h1[nt] = zero8();
#pragma unroll
    for (int ks = 0; ks < 4; ++ks) {
        const int kk = ks * 4 + half * 2;
        v2f a; a.x = stage[wave][q][kk]; a.y = stage[wave][q][kk + 1];
#pragma unroll
        for (int nt = 0; nt < 4; ++nt) {
            v2f b;
            b.x = (kk     < 15) ? rgb_w1[kk * 64 + nt * 16 + q]       : 0.0f;
            b.y = (kk + 1 < 15) ? rgb_w1[(kk + 1) * 64 + nt * 16 + q] : 0.0f;
            h1[nt] = wmma4(a, b, h1[nt]);
        }
    }
    __syncthreads();
#pragma unroll
    for (int nt = 0; nt < 4; ++nt)
#pragma unroll
        for (int i = 0; i < 8; ++i)
            stage[wave][i + half * 8][nt * 16 + q] = fmaxf(h1[nt][i], 0.0f);
    __syncthreads();

    // ---------------- stage 4: h2 = relu(h1(16x64) @ rgb_w2(64x64)) -------
    v8f h2[4];
#pragma unroll
    for (int nt = 0; nt < 4; ++nt) h2[nt] = zero8();
#pragma unroll
    for (int ks = 0; ks < 16; ++ks) {
        const int kk = ks * 4 + half * 2;
        v2f a; a.x = stage[wave][q][kk]; a.y = stage[wave][q][kk + 1];
#pragma unroll
        for (int nt = 0; nt < 4; ++nt) {
            v2f b;
            b.x = rgb_w2[kk * 64 + nt * 16 + q];
            b.y = rgb_w2[(kk + 1) * 64 + nt * 16 + q];
            h2[nt] = wmma4(a, b, h2[nt]);
        }
    }
    __syncthreads();
#pragma unroll
    for (int nt = 0; nt < 4; ++nt)
#pragma unroll
        for (int i = 0; i < 8; ++i)
            stage[wave][i + half * 8][nt * 16 + q] = fmaxf(h2[nt][i], 0.0f);
    __syncthreads();

    // ---------------- stage 5: rgb = sigmoid(h2(16x64) @ rgb_w3(64x3)) ----
    v8f rgbacc = zero8();
#pragma unroll
    for (int ks = 0; ks < 16; ++ks) {
        const int kk = ks * 4 + half * 2;
        v2f a; a.x = stage[wave][q][kk]; a.y = stage[wave][q][kk + 1];
        v2f b;
        b.x = (q < 3) ? rgb_w3[kk * 3 + q]       : 0.0f;
        b.y = (q < 3) ? rgb_w3[(kk + 1) * 3 + q] : 0.0f;
        rgbacc = wmma4(a, b, rgbacc);
    }
    if (q < 3) {
#pragma unroll
        for (int i = 0; i < 8; ++i) {
            const int m = i + half * 8;
            const float s = 1.0f / (1.0f + expf(-rgbacc[i]));
            out[(size_t)(tileBase + m) * 4 + 1 + q] = s;
        }
    }
}

// ---------------------------------------------------------------------------
// Kernel 2: volume-rendering weights. One wave32 per ray (S=128 samples,
// 4 samples/lane). Exclusive cumsum via lane-local scan + __shfl_up wave scan.
// Reads log-density from out[...,0], overwrites it with alpha*trans.
// ---------------------------------------------------------------------------
__global__ __launch_bounds__(256) void volrend_kernel(
    const float* __restrict__ delta,  // (R,S)
    float* __restrict__ out,          // (R,S,4)
    float offset, int R, int S)
{
    const int wave = threadIdx.x >> 5;
    const int lane = threadIdx.x & 31;
    const int ray  = blockIdx.x * (blockDim.x >> 5) + wave;
    if (ray >= R) return;

    const int sBase = lane * 4;  // 32 lanes * 4 = 128 samples
    float dd[4], p[4];
    float run = 0.0f;
#pragma unroll
    for (int j = 0; j < 4; ++j) {
        const size_t pi = (size_t)ray * S + (sBase + j);
        const float ld = out[pi * 4 + 0];
        const float de = delta[pi];
        dd[j] = expf(ld + offset) * de;  // exp(ld + OFFSET + log(delta))
        run += dd[j];
        p[j] = run;
    }
    // inclusive wave scan of per-lane sums (wave32)
    const float laneSum = run;
    float incl = laneSum;
#pragma unroll
    for (int off = 1; off < 32; off <<= 1) {
        const float v = __shfl_up(incl, off, 32);
        if (lane >= off) incl += v;
    }
    const float excl = incl - laneSum;
#pragma unroll
    for (int j = 0; j < 4; ++j) {
        const float e = excl + (j == 0 ? 0.0f : p[j - 1]);
        const float trans = expf(-e);
        const float alpha = 1.0f - expf(-dd[j]);
        const size_t pi = (size_t)ray * S + (sBase + j);
        out[pi * 4 + 0] = alpha * trans;
    }
}

// ---------------------------------------------------------------------------
extern "C" void kernel_launch(void* const* d_in, const int* in_sizes, int n_in,
                              void* d_out, int out_size, void* d_ws, size_t ws_size,
                              hipStream_t stream) {
    const float* xyz    = (const float*)d_in[0];
    const float* delta  = (const float*)d_in[1];
    const float* table  = (const float*)d_in[2];
    const float* w_in   = (const float*)d_in[3];
    const float* w_out  = (const float*)d_in[4];
    const float* rgb_w1 = (const float*)d_in[5];
    const float* rgb_w2 = (const float*)d_in[6];
    const float* rgb_w3 = (const float*)d_in[7];
    float* out = (float*)d_out;

    const int N = in_sizes[1];   // R*S points (delta element count)
    const int S = 128;
    const int R = N / S;

    // 16 points per wave, 8 waves per block
    const int tiles  = N / 16;
    const int blocks = tiles / WAVES_PER_BLOCK;
    ngp_fused_kernel<<<blocks, 256, 0, stream>>>(xyz, table, w_in, w_out,
                                                 rgb_w1, rgb_w2, rgb_w3, out, N);

    // OFFSET = log(log(1/T_THRESH)) - log(FAR-NEAR) - TAU^2/2
    const float offset = (float)(log(log(1.0 / 0.99)) - log(6.0 - 2.0) - 0.5);
    volrend_kernel<<<(R + 7) / 8, 256, 0, stream>>>(delta, out, offset, R, S);
}